// RelGraphConvBDD_86938728005791
// MI455X (gfx1250) — compile-verified
//
#include <hip/hip_runtime.h>

#define N_NODES   100000
#define N_EDGES   1000000
#define IN_FEAT   128
#define OUT_FEAT  128
#define NUM_RELS  64
#define NUM_BASES 16
// SUBMAT_IN = SUBMAT_OUT = 8

typedef float v2f __attribute__((ext_vector_type(2)));
typedef float v8f __attribute__((ext_vector_type(8)));

// ---------------------------------------------------------------------------
// Kernel 0: zero the accumulator (d_out doubles as agg buffer).
// ---------------------------------------------------------------------------
__global__ void zero_kernel(float4* __restrict__ out, long n4) {
    long i = (long)blockIdx.x * blockDim.x + threadIdx.x;
    long stride = (long)gridDim.x * blockDim.x;
    float4 z = make_float4(0.f, 0.f, 0.f, 0.f);
    for (; i < n4; i += stride) out[i] = z;
}

// ---------------------------------------------------------------------------
// Kernel 1: edge messages, one thread per (edge, base).
//   msg[b*8+o] = norm[e] * sum_i h[src[e]][b*8+i] * W[r[e]][b][i][o]
// scatter-summed into out[dst[e]] with f32 atomics (agg fits in 192MB L2).
// Thread t -> edge = t>>4, base = t&15: h gathers and weight reads are
// contiguous float4s; atomic destinations coalesce across the 16 threads
// of one edge.
// ---------------------------------------------------------------------------
__global__ void __launch_bounds__(256) edge_msg_kernel(
    const float*  __restrict__ h,
    const float*  __restrict__ norm,
    const float*  __restrict__ weight,
    const int*    __restrict__ src,
    const int*    __restrict__ dst,
    const int*    __restrict__ rel,
    float*        __restrict__ out)
{
    long tid = (long)blockIdx.x * blockDim.x + threadIdx.x;
    if (tid >= (long)N_EDGES * NUM_BASES) return;
    int e = (int)(tid >> 4);
    int b = (int)(tid & 15);

    int s  = src[e];
    int d  = dst[e];
    int rr = rel[e];
    float nrm = norm[e];

    // 8 source features of this base (two float4 loads, 32B contiguous)
    const float4* hp = (const float4*)(h + (size_t)s * IN_FEAT + b * 8);
    float4 h0 = hp[0];
    float4 h1 = hp[1];
    float hv[8] = {h0.x, h0.y, h0.z, h0.w, h1.x, h1.y, h1.z, h1.w};

    // 64 weights of this base (16 float4 loads, 256B contiguous)
    const float4* wp = (const float4*)(weight + (size_t)rr * 1024 + b * 64);

    float acc[8];
#pragma unroll
    for (int o = 0; o < 8; ++o) acc[o] = 0.f;

#pragma unroll
    for (int i = 0; i < 8; ++i) {
        float4 wa = wp[2 * i];
        float4 wb = wp[2 * i + 1];
        float x = hv[i];
        acc[0] = fmaf(x, wa.x, acc[0]);
        acc[1] = fmaf(x, wa.y, acc[1]);
        acc[2] = fmaf(x, wa.z, acc[2]);
        acc[3] = fmaf(x, wa.w, acc[3]);
        acc[4] = fmaf(x, wb.x, acc[4]);
        acc[5] = fmaf(x, wb.y, acc[5]);
        acc[6] = fmaf(x, wb.z, acc[6]);
        acc[7] = fmaf(x, wb.w, acc[7]);
    }

    float* op = out + (size_t)d * OUT_FEAT + b * 8;
#pragma unroll
    for (int o = 0; o < 8; ++o) atomicAdd(op + o, acc[o] * nrm);
}

// ---------------------------------------------------------------------------
// Kernel 2: self-loop GEMM via V_WMMA_F32_16X16X4_F32 + epilogue.
//   out = relu(agg + bias + h @ loop_weight)
// One wave per 16-row M-tile (100000 = 6250*16 exactly -> no tail, EXEC
// stays all-1s as WMMA requires). Each wave computes a full 16x128 strip
// (8 accumulator tiles, 64 VGPRs of C).
//
// loop_weight is staged into LDS TRANSPOSED: sWt[col*KSTRIDE + k].
// Each lane's B operand (rows krow,krow+1 at one column) is then a single
// 8-byte contiguous ds_load_b64 straight into an even VGPR pair -> no
// repacking movs. KSTRIDE=132 (even, so b64-aligned): read banks per b64 =
// {4i,4i+1} for lanes 0-15 and {4i+2,4i+3} for lanes 16-31 -> all 64 banks
// exactly once, conflict-free.
// ---------------------------------------------------------------------------
#define KSTRIDE 132

__global__ void __launch_bounds__(256) selfloop_wmma_kernel(
    const float* __restrict__ h,
    const float* __restrict__ Wl,     // loop_weight, [K=128][N=128] row major
    const float* __restrict__ bias,   // [128]
    float*       __restrict__ out)    // in: agg, out: final
{
    __shared__ float sWt[OUT_FEAT * KSTRIDE];   // transposed: [col][k]

    // Stage loop_weight into LDS transposed. Thread loads float4 of a row
    // (contiguous global), scatters 4 words column-major into LDS.
    for (int idx = threadIdx.x; idx < IN_FEAT * 32; idx += 256) {
        int row = idx >> 5;                       // k
        int c4  = idx & 31;                       // col/4
        float4 v = ((const float4*)(Wl + row * OUT_FEAT))[c4];
        int col = c4 * 4;
        sWt[(col + 0) * KSTRIDE + row] = v.x;
        sWt[(col + 1) * KSTRIDE + row] = v.y;
        sWt[(col + 2) * KSTRIDE + row] = v.z;
        sWt[(col + 3) * KSTRIDE + row] = v.w;
    }
    __syncthreads();

    const int lane  = threadIdx.x & 31;
    const int wave  = threadIdx.x >> 5;
    const int tile  = blockIdx.x * 8 + wave;        // M-tile index
    const int mtiles = N_NODES / 16;                // 6250 exactly
    if (tile >= mtiles) return;                     // uniform per wave

    const int row0  = tile * 16;
    const int mlane = lane & 15;
    const int khalf = lane >> 4;                    // 0: K{0,1}, 1: K{2,3}

    // C init: agg (current d_out) + bias. C elem (v,lane) = (M=v+8*khalf, N=mlane)
    v8f c[8];
#pragma unroll
    for (int n = 0; n < 8; ++n) {
        int col = n * 16 + mlane;
        float bv = bias[col];
#pragma unroll
        for (int v = 0; v < 8; ++v) {
            int m = row0 + v + 8 * khalf;
            c[n][v] = out[(size_t)m * OUT_FEAT + col] + bv;
        }
    }

    // A: lane holds row (row0 + mlane), K-pair (khalf*2, khalf*2+1): b64 loads.
    const float* ap = h + (size_t)(row0 + mlane) * IN_FEAT + khalf * 2;

    for (int k0 = 0; k0 < IN_FEAT; k0 += 4) {
        v2f a = *(const v2f*)(ap + k0);             // A[m][k0+2h], A[m][k0+2h+1]
        int krow = k0 + 2 * khalf;
#pragma unroll
        for (int n = 0; n < 8; ++n) {
            int col = n * 16 + mlane;
            // B[krow][col], B[krow+1][col] -> one ds_load_b64
            v2f b = *(const v2f*)(&sWt[col * KSTRIDE + krow]);
            c[n] = __builtin_amdgcn_wmma_f32_16x16x4_f32(
                false, a, false, b, (short)0, c[n], false, false);
        }
    }

    // Epilogue: ReLU + store.
#pragma unroll
    for (int n = 0; n < 8; ++n) {
        int col = n * 16 + mlane;
#pragma unroll
        for (int v = 0; v < 8; ++v) {
            int m = row0 + v + 8 * khalf;
            out[(size_t)m * OUT_FEAT + col] = fmaxf(c[n][v], 0.f);
        }
    }
}

// ---------------------------------------------------------------------------
// Launcher.  Inputs (setup_inputs order):
//  0 h(f32 N*128)  1 norm(f32 E)  2 weight(f32 64*1024)  3 loop_weight(f32 128*128)
//  4 h_bias(f32 128)  5 src(i32 E)  6 dst(i32 E)  7 r(i32 E)
// ---------------------------------------------------------------------------
extern "C" void kernel_launch(void* const* d_in, const int* in_sizes, int n_in,
                              void* d_out, int out_size, void* d_ws, size_t ws_size,
                              hipStream_t stream) {
    const float* h    = (const float*)d_in[0];
    const float* norm = (const float*)d_in[1];
    const float* wgt  = (const float*)d_in[2];
    const float* wl   = (const float*)d_in[3];
    const float* bias = (const float*)d_in[4];
    const int*   src  = (const int*)d_in[5];
    const int*   dst  = (const int*)d_in[6];
    const int*   rel  = (const int*)d_in[7];
    float* out = (float*)d_out;

    // 1) zero agg
    long n4 = (long)N_NODES * OUT_FEAT / 4;
    zero_kernel<<<1024, 256, 0, stream>>>((float4*)out, n4);

    // 2) edge messages + scatter
    long ethreads = (long)N_EDGES * NUM_BASES;
    int eblocks = (int)((ethreads + 255) / 256);
    edge_msg_kernel<<<eblocks, 256, 0, stream>>>(h, norm, wgt, src, dst, rel, out);

    // 3) self-loop WMMA GEMM + bias + relu (in place over agg)
    int gblocks = (N_NODES / 16 + 7) / 8;   // 6250 wave-tiles / 8 waves per block
    selfloop_wmma_kernel<<<gblocks, 256, 0, stream>>>(h, wl, bias, out);
}